// GraphSelfAttention_1726576856579
// MI455X (gfx1250) — compile-verified
//
#include <hip/hip_runtime.h>
#include <math.h>

#if defined(__HIP_DEVICE_COMPILE__)
#if !__has_builtin(__builtin_amdgcn_wmma_f32_16x16x4_f32)
#error "wmma_f32_16x16x4_f32 builtin not available for this offload target"
#endif
#if __has_builtin(__builtin_amdgcn_global_load_async_to_lds_b128)
#define HAVE_ASYNC_LDS 1
#endif
#endif

typedef __attribute__((ext_vector_type(2))) float v2f;
typedef __attribute__((ext_vector_type(4))) float v4f;
typedef __attribute__((ext_vector_type(8))) float v8f;

#if defined(HAVE_ASYNC_LDS)
// Builtin signature (from compiler diagnostic): parameters are generic
// pointers of type 'int __attribute__((vector_size(16))) *'.
typedef int v4i_vs __attribute__((vector_size(16)));
// global_load_async_to_lds_b128: 16B per lane, Memory -> LDS, ASYNCcnt-tracked
#define ASYNC_CP16(gp, lp)                                                  \
  __builtin_amdgcn_global_load_async_to_lds_b128(                           \
      (v4i_vs*)(gp), (v4i_vs*)(lp), 0, 0)
__device__ __forceinline__ void wait_async_then_barrier() {
#if __has_builtin(__builtin_amdgcn_s_wait_asynccnt)
  __builtin_amdgcn_s_wait_asynccnt(0);
#else
  asm volatile("s_wait_asynccnt 0x0" ::: "memory");
#endif
  __syncthreads();
}
#endif

#define DM     512
#define S_LEN  5440            // 4096 + 1024 + 256 + 64
#define BATCH  8
#define ROWS   (BATCH * S_LEN) // 43520
#define NH     8
#define DK     64
#define MA     10              // max_attn for (4096, win=5, stride=4)

#define SROW   20              // padded LDS row stride (floats): 80B, 16B-aligned,
                               // n*20 mod 64 hits 16 distinct banks -> conflict-free

// ------------------------------------------------------------------
// Kernel 1: fused QKV projection.  out[r][n] = sum_d x[r][d]*W[n][d]
// (q scaled by 1/sqrt(64)).  Tile 128x64, 8 waves, fp32 WMMA 16x16x4.
// A and B both stored row-major in LDS (Bs[n][k]) so each B fragment
// is one ds_load_2addr_b32 into an adjacent VGPR pair.
// ------------------------------------------------------------------
#define K1_RT 128
#define K1_NT 64
#define K1_KT 16

__global__ __launch_bounds__(256) void qkv_gemm(
    const float* __restrict__ x,
    const float* __restrict__ Wq, const float* __restrict__ Wk,
    const float* __restrict__ Wv,
    float* __restrict__ qo, float* __restrict__ ko, float* __restrict__ vo)
{
  __shared__ float As[K1_RT * SROW];   // 128 x 16 (padded to 20)
  __shared__ float Bs[K1_NT * SROW];   // 64 x 16  (padded to 20)

  const int tid  = threadIdx.x;
  const int lane = tid & 31;
  const int wave = tid >> 5;
  const int r0   = blockIdx.x * K1_RT;
  const int n0g  = blockIdx.y * K1_NT;     // 0..1535 in 64-steps
  const int which = n0g / DM;
  const int n0   = n0g - which * DM;
  const float* W = (which == 0) ? Wq : (which == 1) ? Wk : Wv;
  float* OUT     = (which == 0) ? qo : (which == 1) ? ko : vo;
  const float scale = (which == 0) ? 0.125f : 1.0f;  // 1/sqrt(D_K)

  const int l15 = lane & 15;
  const int hi  = lane >> 4;

  v8f acc[4];
  #pragma unroll
  for (int s = 0; s < 4; ++s)
    #pragma unroll
    for (int j = 0; j < 8; ++j) acc[s][j] = 0.0f;

  for (int k0g = 0; k0g < DM; k0g += K1_KT) {
#if defined(HAVE_ASYNC_LDS)
    // A tile: 128 rows x 4 x 16B segments = 512 copies, 2 per thread
    #pragma unroll
    for (int it = 0; it < 2; ++it) {
      int idx = tid + it * 256;
      int row = idx >> 2, seg = idx & 3;
      ASYNC_CP16(x + (size_t)(r0 + row) * DM + k0g + seg * 4,
                 &As[row * SROW + seg * 4]);
    }
    // B tile: 64 rows x 4 segments, 1 per thread
    {
      int n = tid >> 2, seg = tid & 3;
      ASYNC_CP16(W + (size_t)(n0 + n) * DM + k0g + seg * 4,
                 &Bs[n * SROW + seg * 4]);
    }
    wait_async_then_barrier();
#else
    #pragma unroll
    for (int it = 0; it < 2; ++it) {
      int idx = tid + it * 256;
      int row = idx >> 2, seg = idx & 3;
      *(v4f*)&As[row * SROW + seg * 4] =
          *(const v4f*)(x + (size_t)(r0 + row) * DM + k0g + seg * 4);
    }
    {
      int n = tid >> 2, seg = tid & 3;
      *(v4f*)&Bs[n * SROW + seg * 4] =
          *(const v4f*)(W + (size_t)(n0 + n) * DM + k0g + seg * 4);
    }
    if (k0g + K1_KT < DM)
      __builtin_prefetch(x + (size_t)(r0 + (tid >> 2)) * DM + k0g + K1_KT, 0, 0);
    __syncthreads();
#endif

    const int rA = wave * 16;
    #pragma unroll
    for (int k0 = 0; k0 < K1_KT; k0 += 4) {
      // A frag: lanes 0-15 hold K{k0,k0+1}, lanes 16-31 K{k0+2,k0+3} (b64)
      v2f a = *(const v2f*)&As[(rA + l15) * SROW + k0 + 2 * hi];
      #pragma unroll
      for (int s = 0; s < 4; ++s) {
        // B frag: per lane {B[k0+hi][n], B[k0+hi+2][n]} = one ds_load_2addr
        const float* bp = &Bs[(16 * s + l15) * SROW + k0 + hi];
        v2f b; b.x = bp[0]; b.y = bp[2];
        acc[s] = __builtin_amdgcn_wmma_f32_16x16x4_f32(
            false, a, false, b, (short)0, acc[s], false, false);
      }
    }
    __syncthreads();
  }

  // D layout: VGPR j -> row M=j (lanes 0-15) / M=j+8 (lanes 16-31), col=lane&15
  const int rA = wave * 16;
  #pragma unroll
  for (int s = 0; s < 4; ++s)
    #pragma unroll
    for (int j = 0; j < 8; ++j) {
      int row = r0 + rA + j + 8 * hi;
      int col = n0 + 16 * s + l15;
      OUT[(size_t)row * DM + col] = acc[s][j] * scale;
    }
}

// ------------------------------------------------------------------
// Kernel 2: pyramid-graph attention. One wave32 per (b,s,h); each
// lane owns dims {lane, lane+32} of d_k=64. Butterfly reductions.
// Matches reference: invalid scores forced to 0.0 BEFORE softmax
// (stay in denominator), excluded from the numerator only.
// ------------------------------------------------------------------
__global__ __launch_bounds__(256) void pam_attn(
    const float* __restrict__ q, const float* __restrict__ k,
    const float* __restrict__ v, const int* __restrict__ mask,
    float* __restrict__ o)
{
  const int tid  = threadIdx.x;
  const int lane = tid & 31;
  const int wid  = blockIdx.x * 8 + (tid >> 5);  // 0 .. ROWS*NH-1
  const int h  = wid & 7;
  const int bs = wid >> 3;
  const int s  = bs % S_LEN;
  const int b  = bs / S_LEN;

  const size_t qb = (size_t)bs * DM + h * DK;
  const float q0 = q[qb + lane];
  const float q1 = q[qb + lane + 32];

  float sc[MA];
  int   idxm[MA];
  unsigned vbits = 0;
  #pragma unroll
  for (int m = 0; m < MA; ++m) {
    int mi = mask[s * MA + m];
    bool valid = (mi >= 0);
    int safe = valid ? mi : 0;
    idxm[m] = safe;
    if (valid) vbits |= (1u << m);
    size_t kb = ((size_t)b * S_LEN + safe) * DM + h * DK;
    float p = q0 * k[kb + lane] + q1 * k[kb + lane + 32];
    #pragma unroll
    for (int off = 16; off > 0; off >>= 1) p += __shfl_xor(p, off, 32);
    sc[m] = valid ? p : 0.0f;
  }

  float mx = sc[0];
  #pragma unroll
  for (int m = 1; m < MA; ++m) mx = fmaxf(mx, sc[m]);
  float den = 0.0f;
  #pragma unroll
  for (int m = 0; m < MA; ++m) { sc[m] = __expf(sc[m] - mx); den += sc[m]; }
  const float inv = 1.0f / den;

  float o0 = 0.0f, o1 = 0.0f;
  #pragma unroll
  for (int m = 0; m < MA; ++m) {
    if (vbits & (1u << m)) {
      size_t vb = ((size_t)b * S_LEN + idxm[m]) * DM + h * DK;
      float w = sc[m];
      o0 += w * v[vb + lane];
      o1 += w * v[vb + lane + 32];
    }
  }
  o[qb + lane]      = o0 * inv;
  o[qb + lane + 32] = o1 * inv;
}

// ------------------------------------------------------------------
// Kernel 3: ctx = attn @ Wfc^T + bfc + x, then LayerNorm, fused.
// Block owns 16 rows x ALL 512 cols (LN sees full rows). 8 waves,
// each wave = 64-col strip (4 WMMA subtiles). C-tile aliases the
// staging LDS to stay under the 64KB static-LDS ceiling.
// ------------------------------------------------------------------
#define K3_RT 16
#define K3_KT 16
#define K3_SC 520   // padded Cs stride over n

__global__ __launch_bounds__(256) void fc_res_ln(
    const float* __restrict__ att, const float* __restrict__ Wfc,
    const float* __restrict__ bfc, const float* __restrict__ x,
    const float* __restrict__ gamma, const float* __restrict__ beta,
    float* __restrict__ out)
{
  __shared__ float smem[K3_RT * SROW + DM * SROW];  // 10560 f = 42.2KB
  float* As = smem;                 // [16][20]
  float* Bs = smem + K3_RT * SROW;  // [512][20]  (Wfc rows, untransposed)

  const int tid  = threadIdx.x;
  const int lane = tid & 31;
  const int wave = tid >> 5;
  const int r0   = blockIdx.x * K3_RT;
  const int l15  = lane & 15;
  const int hi   = lane >> 4;
  const int nc0  = wave * 64;

  v8f acc[4];
  #pragma unroll
  for (int s = 0; s < 4; ++s)
    #pragma unroll
    for (int j = 0; j < 8; ++j) acc[s][j] = 0.0f;

  for (int k0g = 0; k0g < DM; k0g += K3_KT) {
#if defined(HAVE_ASYNC_LDS)
    if (tid < 64) {   // A tile: 16 rows x 4 segments
      int row = tid >> 2, seg = tid & 3;
      ASYNC_CP16(att + (size_t)(r0 + row) * DM + k0g + seg * 4,
                 &As[row * SROW + seg * 4]);
    }
    #pragma unroll
    for (int it = 0; it < 8; ++it) {  // B tile: 512 rows x 4 segments
      int idx = tid + it * 256;
      int n = idx >> 2, seg = idx & 3;
      ASYNC_CP16(Wfc + (size_t)n * DM + k0g + seg * 4,
                 &Bs[n * SROW + seg * 4]);
    }
    wait_async_then_barrier();
#else
    if (tid < 64) {
      int row = tid >> 2, seg = tid & 3;
      *(v4f*)&As[row * SROW + seg * 4] =
          *(const v4f*)(att + (size_t)(r0 + row) * DM + k0g + seg * 4);
    }
    #pragma unroll
    for (int it = 0; it < 8; ++it) {
      int idx = tid + it * 256;
      int n = idx >> 2, seg = idx & 3;
      *(v4f*)&Bs[n * SROW + seg * 4] =
          *(const v4f*)(Wfc + (size_t)n * DM + k0g + seg * 4);
    }
    __syncthreads();
#endif

    #pragma unroll
    for (int k0 = 0; k0 < K3_KT; k0 += 4) {
      v2f a = *(const v2f*)&As[l15 * SROW + k0 + 2 * hi];
      #pragma unroll
      for (int s = 0; s < 4; ++s) {
        const float* bp = &Bs[(nc0 + 16 * s + l15) * SROW + k0 + hi];
        v2f b; b.x = bp[0]; b.y = bp[2];
        acc[s] = __builtin_amdgcn_wmma_f32_16x16x4_f32(
            false, a, false, b, (short)0, acc[s], false, false);
      }
    }
    __syncthreads();
  }

  // Epilogue: C tile (aliases staging LDS; all As/Bs reads complete)
  float* Cs = smem;   // [16][520] = 8320 floats <= 10560 available
  #pragma unroll
  for (int s = 0; s < 4; ++s)
    #pragma unroll
    for (int j = 0; j < 8; ++j) {
      int row = j + 8 * hi;
      int col = nc0 + 16 * s + l15;
      Cs[row * K3_SC + col] =
          acc[s][j] + bfc[col] + x[(size_t)(r0 + row) * DM + col];
    }
  __syncthreads();

  // LayerNorm: each wave normalizes 2 rows (512 elems, 16/lane)
  #pragma unroll
  for (int rr = 0; rr < 2; ++rr) {
    int row = wave * 2 + rr;
    float sum = 0.0f, sq = 0.0f;
    #pragma unroll
    for (int i = 0; i < 16; ++i) {
      float c = Cs[row * K3_SC + lane + 32 * i];
      sum += c; sq += c * c;
    }
    #pragma unroll
    for (int off = 16; off > 0; off >>= 1) {
      sum += __shfl_xor(sum, off, 32);
      sq  += __shfl_xor(sq,  off, 32);
    }
    float mu  = sum * (1.0f / 512.0f);
    float var = sq  * (1.0f / 512.0f) - mu * mu;
    float rs  = rsqrtf(var + 1e-6f);
    #pragma unroll
    for (int i = 0; i < 16; ++i) {
      int c = lane + 32 * i;
      float vl = Cs[row * K3_SC + c];
      out[(size_t)(r0 + row) * DM + c] = (vl - mu) * rs * gamma[c] + beta[c];
    }
  }
}

// ------------------------------------------------------------------
extern "C" void kernel_launch(void* const* d_in, const int* in_sizes, int n_in,
                              void* d_out, int out_size, void* d_ws, size_t ws_size,
                              hipStream_t stream)
{
  const float* x     = (const float*)d_in[0];
  const float* Wq    = (const float*)d_in[1];
  const float* Wk    = (const float*)d_in[2];
  const float* Wv    = (const float*)d_in[3];
  const float* Wfc   = (const float*)d_in[4];
  const float* bfc   = (const float*)d_in[5];
  const float* gamma = (const float*)d_in[6];
  const float* beta  = (const float*)d_in[7];
  const int*   mask  = (const int*)d_in[8];
  float* out = (float*)d_out;

  float* ws = (float*)d_ws;
  const size_t PLANE = (size_t)ROWS * DM;   // 43520*512 floats
  float* q   = ws;
  float* k   = ws + PLANE;
  float* v   = ws + 2 * PLANE;
  float* att = ws + 3 * PLANE;

  qkv_gemm<<<dim3(ROWS / K1_RT, (3 * DM) / K1_NT), 256, 0, stream>>>(
      x, Wq, Wk, Wv, q, k, v);
  pam_attn<<<dim3(ROWS), 256, 0, stream>>>(q, k, v, mask, att);
  fc_res_ln<<<dim3(ROWS / K3_RT), 256, 0, stream>>>(
      att, Wfc, bfc, x, gamma, beta, out);
}